// DiffusionProcess_21698174780217
// MI455X (gfx1250) — compile-verified
//
#include <hip/hip_runtime.h>
#include <stdint.h>

// ---------------------------------------------------------------------------
// DiffusionProcess fused kernel for MI455X (gfx1250, wave32).
//
// out[i] = x[i] + sum_s sqrt(beta_s) * z_s[i],  z_s = JAX threefry normal.
// Single pass over x (~100 MiB traffic, ~4.3 us floor @ 23.3 TB/s); the real
// cost is ~6e10 VALU ops of Threefry + erfinv, so the kernel is built for
// VALU throughput: 4 independent RNG streams per thread (ILP), step constants
// broadcast from LDS as one ds_load_b128 per step, fast __logf erfinv.
//
// CDNA5 path: async Global->LDS copy (global_load_async_to_lds_b32 +
// s_wait_asynccnt) stages the 100-step {k0,k1,k2,sqrt(2*beta)} table.
// ---------------------------------------------------------------------------

#define ROTL32(v, r) (((v) << (r)) | ((v) >> (32 - (r))))

// Threefry-2x32 with precomputed k2 (JAX key schedule + rotation constants).
__device__ __forceinline__ void tf2x32_k(uint32_t k0, uint32_t k1, uint32_t k2,
                                         uint32_t x0, uint32_t x1,
                                         uint32_t& o0, uint32_t& o1) {
  uint32_t X0 = x0 + k0;
  uint32_t X1 = x1 + k1;
#define TF_R(r) X0 += X1; X1 = ROTL32(X1, r); X1 ^= X0;
  TF_R(13) TF_R(15) TF_R(26) TF_R(6)  X0 += k1; X1 += k2 + 1u;
  TF_R(17) TF_R(29) TF_R(16) TF_R(24) X0 += k2; X1 += k0 + 2u;
  TF_R(13) TF_R(15) TF_R(26) TF_R(6)  X0 += k0; X1 += k1 + 3u;
  TF_R(17) TF_R(29) TF_R(16) TF_R(24) X0 += k1; X1 += k2 + 4u;
  TF_R(13) TF_R(15) TF_R(26) TF_R(6)  X0 += k2; X1 += k0 + 5u;
#undef TF_R
  o0 = X0; o1 = X1;
}

__device__ __forceinline__ void tf2x32(uint32_t k0, uint32_t k1,
                                       uint32_t x0, uint32_t x1,
                                       uint32_t& o0, uint32_t& o1) {
  tf2x32_k(k0, k1, k0 ^ k1 ^ 0x1BD11BDAu, x0, x1, o0, o1);
}

// One word of jax.random.split(key(1), n): flat = concat(y0[0:n], y1[0:n]) of
// blocks (i, n+i); keys[s] = (flat[2s], flat[2s+1]).
__device__ __forceinline__ uint32_t split_word(uint32_t n, uint32_t idx) {
  uint32_t y0, y1;
  if (idx < n) { tf2x32(0u, 1u, idx, n + idx, y0, y1); return y0; }
  tf2x32(0u, 1u, idx - n, idx, y0, y1); return y1;
}

// bits -> uniform in [-0.99999994, 1.0)  (JAX: mantissa trick, scale, clamp)
__device__ __forceinline__ float bits_to_u(uint32_t b) {
  const float lo = -0.99999994f;
  const float span = 1.0f - lo;
  float f = __uint_as_float((b >> 9) | 0x3f800000u) - 1.0f;
  float u = f * span + lo;
  return fmaxf(u, lo);
}

// Single-precision erfinv (Giles 2010), matches XLA's f32 ErfInv to ~ulp.
// Tail branch (|x| > ~0.9966, ~0.3% of samples) is rarely active per wave.
__device__ __forceinline__ float erfinv_fast(float x) {
  float w = -__logf((1.0f - x) * (1.0f + x));
  float p;
  if (w < 5.0f) {
    w = w - 2.5f;
    p =               2.81022636e-08f;
    p = fmaf(p, w,    3.43273939e-07f);
    p = fmaf(p, w,   -3.52338770e-06f);
    p = fmaf(p, w,   -4.39150654e-06f);
    p = fmaf(p, w,    2.18580870e-04f);
    p = fmaf(p, w,   -1.25372503e-03f);
    p = fmaf(p, w,   -4.17768164e-03f);
    p = fmaf(p, w,    2.46640727e-01f);
    p = fmaf(p, w,    1.50140941e+00f);
  } else {
    w = sqrtf(w) - 3.0f;
    p =              -2.00214257e-04f;
    p = fmaf(p, w,    1.00950558e-04f);
    p = fmaf(p, w,    1.34934322e-03f);
    p = fmaf(p, w,   -3.67342844e-03f);
    p = fmaf(p, w,    5.73950773e-03f);
    p = fmaf(p, w,   -7.62246130e-03f);
    p = fmaf(p, w,    9.43887047e-03f);
    p = fmaf(p, w,    1.00167406e+00f);
    p = fmaf(p, w,    2.83297682e+00f);
  }
  return x * p;
}

// ---------------------------------------------------------------------------
// Setup: derive the 100 step keys (exact jax.random.split semantics), the
// Threefry k2, and c = sqrt(2*beta).  ws layout: interleaved 16B records
//   ws[4s+0]=k0, ws[4s+1]=k1, ws[4s+2]=k2, ws[4s+3]=bitcast(sqrt(2)*sqrt(beta))
// ---------------------------------------------------------------------------
__global__ void dp_setup(const float* __restrict__ betas,
                         uint32_t* __restrict__ ws, int n) {
  const int s = blockIdx.x * blockDim.x + threadIdx.x;
  if (s >= n) return;
  const uint32_t k0 = split_word((uint32_t)n, (uint32_t)(2 * s));
  const uint32_t k1 = split_word((uint32_t)n, (uint32_t)(2 * s + 1));
  ws[4 * s + 0] = k0;
  ws[4 * s + 1] = k1;
  ws[4 * s + 2] = k0 ^ k1 ^ 0x1BD11BDAu;
  ws[4 * s + 3] = __float_as_uint(sqrtf(betas[s]) * 1.41421356237f);
}

// ---------------------------------------------------------------------------
// Main fused kernel. Thread t owns threefry blocks {t, t+quarter}, i.e.
// elements {t, t+halfN, t+quarter, t+quarter+halfN} (JAX half-split layout).
// ---------------------------------------------------------------------------
__global__ __launch_bounds__(256) void dp_diffuse(const float* __restrict__ x,
                                                  const uint32_t* __restrict__ ws,
                                                  float* __restrict__ out,
                                                  int nsteps, int quarter,
                                                  int halfN) {
  __shared__ __attribute__((aligned(16))) uint32_t sk[400];  // 100 x {k0,k1,k2,c}

  // --- CDNA5 async Global->LDS stage of the step-constant table ------------
  {
    const int total = 4 * nsteps;  // 400 dwords
    const uint64_t base = (uint64_t)(uintptr_t)ws;
    for (int ofs = threadIdx.x; ofs < total; ofs += 256) {
      uint32_t lds  = (uint32_t)(uintptr_t)(&sk[ofs]);  // low 32 bits = LDS addr
      uint32_t voff = (uint32_t)(ofs * 4);
      asm volatile("global_load_async_to_lds_b32 %0, %1, %2"
                   :: "v"(lds), "v"(voff), "s"(base) : "memory");
    }
    asm volatile("s_wait_asynccnt 0" ::: "memory");
  }
  __syncthreads();

  const int t = blockIdx.x * 256 + threadIdx.x;
  if (t >= quarter) return;

  const int b0 = t;
  const int b1 = t + quarter;

  // One read of x per element (four coalesced streams).
  const float xv0 = x[b0];
  const float xv1 = x[b0 + halfN];
  const float xv2 = x[b1];
  const float xv3 = x[b1 + halfN];

  float a0 = 0.0f, a1 = 0.0f, a2 = 0.0f, a3 = 0.0f;

#pragma unroll 1
  for (int s = 0; s < nsteps; ++s) {
    // Single ds_load_b128 broadcast of this step's constants.
    const uint4 kc = *(const uint4*)(&sk[4 * s]);
    const uint32_t k0 = kc.x;
    const uint32_t k1 = kc.y;
    const uint32_t k2 = kc.z;
    const float    c  = __uint_as_float(kc.w);  // sqrt(2*beta_s)

    uint32_t r0, r1, r2, r3;
    tf2x32_k(k0, k1, k2, (uint32_t)b0, (uint32_t)(b0 + halfN), r0, r1);
    tf2x32_k(k0, k1, k2, (uint32_t)b1, (uint32_t)(b1 + halfN), r2, r3);

    a0 = fmaf(c, erfinv_fast(bits_to_u(r0)), a0);
    a1 = fmaf(c, erfinv_fast(bits_to_u(r1)), a1);
    a2 = fmaf(c, erfinv_fast(bits_to_u(r2)), a2);
    a3 = fmaf(c, erfinv_fast(bits_to_u(r3)), a3);
  }

  out[b0]          = xv0 + a0;
  out[b0 + halfN]  = xv1 + a1;
  out[b1]          = xv2 + a2;
  out[b1 + halfN]  = xv3 + a3;
}

// ---------------------------------------------------------------------------
extern "C" void kernel_launch(void* const* d_in, const int* in_sizes, int n_in,
                              void* d_out, int out_size, void* d_ws, size_t ws_size,
                              hipStream_t stream) {
  const float* x     = (const float*)d_in[0];
  const float* betas = (const float*)d_in[1];
  float*       out   = (float*)d_out;
  uint32_t*    ws    = (uint32_t*)d_ws;

  const int nsteps  = in_sizes[1];   // 100
  const int N       = in_sizes[0];   // 64*3*256*256 = 12,582,912
  const int halfN   = N / 2;
  const int quarter = N / 4;

  dp_setup<<<(nsteps + 127) / 128, 128, 0, stream>>>(betas, ws, nsteps);

  const int threads = 256;
  const int blocks  = (quarter + threads - 1) / threads;
  dp_diffuse<<<blocks, threads, 0, stream>>>(x, ws, out, nsteps, quarter, halfN);
}